// MultidimensionalProcessor_50568944943392
// MI455X (gfx1250) — compile-verified
//
#include <hip/hip_runtime.h>
#include <math.h>

typedef float v2f __attribute__((ext_vector_type(2)));
typedef float v8f __attribute__((ext_vector_type(8)));

#define NP   8
#define NB   4
#define NH   4
#define EDIM 12
#define SEQ  1024
#define NTOK (NB * SEQ)            // 4096 tokens
#define QKVSZ (NP * NTOK * EDIM)   // 393216 floats per q/k/v/o buffer
#define VTS  1032                  // padded V^T row stride (floats)

__device__ __forceinline__ v8f wmma4(v2f a, v2f b, v8f c) {
  // D(16x16,f32) = A(16x4,f32) * B(4x16,f32) + C
  return __builtin_amdgcn_wmma_f32_16x16x4_f32(
      /*neg_a=*/false, a, /*neg_b=*/false, b,
      /*c_mod=*/(short)0, c, /*reuse_a=*/false, /*reuse_b=*/false);
}

// ---------------------------------------------------------------------------
// Kernel A: fold weight chains.
//   Wqkv[which][p][f][e] = in_w[which,f,:] . proj_w[src(p),:,e]   (q scaled by 1/sqrt3)
//   bqkv[which][p][f]    = in_b[which,f] + in_w[which,f,:] . proj_b[src(p),:]
//   FW[p][f][e]          = sum_g fus_w1[f,p*12+g]*out_w[g,e]
//   fb[f]                = fus_b1[f] + sum_p sum_g fus_w1[f,p*12+g]*out_b[g]
// ---------------------------------------------------------------------------
__global__ __launch_bounds__(256) void fold_kernel(
    const float* __restrict__ proj_w, const float* __restrict__ proj_b,
    const float* __restrict__ in_w,   const float* __restrict__ in_b,
    const float* __restrict__ out_w,  const float* __restrict__ out_b,
    const float* __restrict__ fus_w1, const float* __restrict__ fus_b1,
    float* __restrict__ Wqkv, float* __restrict__ bqkv,
    float* __restrict__ FW, float* __restrict__ fb) {
  const int PI[8] = {0, 0, 0, 0, 0, 0, 0, 1};
  const int PJ[8] = {1, 2, 3, 4, 5, 6, 7, 0};
  const float qscale = 0.5773502691896258f;  // 1/sqrt(D=3)
  const int tid = threadIdx.x;

  for (int idx = tid; idx < 3 * NP * EDIM * EDIM; idx += 256) {
    int which = idx / (NP * EDIM * EDIM);
    int rem   = idx % (NP * EDIM * EDIM);
    int p = rem / (EDIM * EDIM);
    int f = (rem / EDIM) % EDIM;
    int e = rem % EDIM;
    int src = (which == 0) ? PI[p] : PJ[p];
    float acc = 0.f;
    for (int g = 0; g < EDIM; ++g)
      acc += in_w[(which * EDIM + f) * EDIM + g] * proj_w[(src * EDIM + g) * EDIM + e];
    if (which == 0) acc *= qscale;
    Wqkv[idx] = acc;
  }
  for (int idx = tid; idx < 3 * NP * EDIM; idx += 256) {
    int which = idx / (NP * EDIM);
    int rem   = idx % (NP * EDIM);
    int p = rem / EDIM;
    int f = rem % EDIM;
    int src = (which == 0) ? PI[p] : PJ[p];
    float acc = in_b[which * EDIM + f];
    for (int g = 0; g < EDIM; ++g)
      acc += in_w[(which * EDIM + f) * EDIM + g] * proj_b[src * EDIM + g];
    if (which == 0) acc *= qscale;
    bqkv[idx] = acc;
  }
  for (int idx = tid; idx < NP * 48 * EDIM; idx += 256) {
    int p = idx / (48 * EDIM);
    int f = (idx / EDIM) % 48;
    int e = idx % EDIM;
    float acc = 0.f;
    for (int g = 0; g < EDIM; ++g)
      acc += fus_w1[f * (NP * EDIM) + p * EDIM + g] * out_w[g * EDIM + e];
    FW[idx] = acc;
  }
  for (int f = tid; f < 48; f += 256) {
    float acc = fus_b1[f];
    for (int p = 0; p < NP; ++p)
      for (int g = 0; g < EDIM; ++g)
        acc += fus_w1[f * (NP * EDIM) + p * EDIM + g] * out_b[g];
    fb[f] = acc;
  }
}

// ---------------------------------------------------------------------------
// Kernel B: q/k/v = x @ Wc^T + bc per pair.  Layout [p][token(=b*S+s)][12].
// ---------------------------------------------------------------------------
__global__ __launch_bounds__(256) void qkv_kernel(
    const float* __restrict__ x, const float* __restrict__ Wqkv,
    const float* __restrict__ bqkv,
    float* __restrict__ q, float* __restrict__ k, float* __restrict__ v) {
  __shared__ float sW[3 * EDIM * EDIM];
  __shared__ float sb[3 * EDIM];
  const int p = blockIdx.y;
  const int tid = threadIdx.x;
  for (int i = tid; i < 3 * EDIM * EDIM; i += 256) {
    int which = i / (EDIM * EDIM);
    int r = i % (EDIM * EDIM);
    sW[i] = Wqkv[which * (NP * EDIM * EDIM) + p * (EDIM * EDIM) + r];
  }
  for (int i = tid; i < 3 * EDIM; i += 256) {
    int which = i / EDIM;
    int f = i % EDIM;
    sb[i] = bqkv[which * (NP * EDIM) + p * EDIM + f];
  }
  __syncthreads();

  const int t = blockIdx.x * 256 + tid;  // 0..4095
  float xv[EDIM];
#pragma unroll
  for (int e = 0; e < EDIM; ++e) xv[e] = x[t * EDIM + e];

  float* outs[3] = {q, k, v};
#pragma unroll
  for (int which = 0; which < 3; ++which) {
#pragma unroll
    for (int f = 0; f < EDIM; ++f) {
      float acc = sb[which * EDIM + f];
#pragma unroll
      for (int e = 0; e < EDIM; ++e)
        acc += xv[e] * sW[which * (EDIM * EDIM) + f * EDIM + e];
      outs[which][p * (NTOK * EDIM) + t * EDIM + f] = acc;
    }
  }
}

// ---------------------------------------------------------------------------
// Kernel C: attention. One 256-thread block (8 waves) per (pair,batch,head);
// each wave owns 8 q-tiles of 16 rows. Scores computed TRANSPOSED
// (S^T = K * Q^T via V_WMMA_F32_16X16X4_F32) so each lane holds one q-row ->
// softmax stats are per-lane. attn@V computed as O^T = V^T * P^T with P^T
// built from registers + 4 packed cross-half shuffles per tile (each
// ds_bpermute carries a different value per half). No in-loop barriers.
// ---------------------------------------------------------------------------
__global__ __launch_bounds__(256) void attn_kernel(
    const float* __restrict__ q, const float* __restrict__ k,
    const float* __restrict__ v, float* __restrict__ o) {
  __shared__ float kb[SEQ * 4];     // K rows padded to 4 (col3 = 0), 16KB
  __shared__ float vt[4 * VTS];     // V transposed, row3 = zeros, 16.5KB

  const int tid  = threadIdx.x;
  const int lane = tid & 31;
  const int wave = tid >> 5;        // 0..7
  const int half = lane >> 4;       // 0: lanes 0-15, 1: lanes 16-31
  const int l16  = lane & 15;       // q-row within tile

  const int bx = blockIdx.x;        // (p*NB + b)*NH + h : 0..127
  const int h  = bx & 3;
  const int pb = bx >> 2;
  const int base = pb * (SEQ * EDIM) + h * 3;   // element (s,d): base + s*12 + d

  // stage K (padded rows) and V (transposed, padded) into LDS
  for (int i = tid; i < SEQ; i += 256) {
    const float* kp = k + base + i * EDIM;
    float4 kv;
    kv.x = kp[0]; kv.y = kp[1]; kv.z = kp[2]; kv.w = 0.f;
    *(float4*)(kb + i * 4) = kv;
    const float* vp = v + base + i * EDIM;
    vt[0 * VTS + i] = vp[0];
    vt[1 * VTS + i] = vp[1];
    vt[2 * VTS + i] = vp[2];
    vt[3 * VTS + i] = 0.f;
  }
  __syncthreads();

#pragma unroll 1
  for (int qi = 0; qi < 8; ++qi) {
    const int qt = wave * 8 + qi;
    // Q tile -> B operand regs (B[d][qrow]); q pre-scaled by 1/sqrt(3)
    const float* qp = q + base + (qt * 16 + l16) * EDIM;
    v2f qreg;
    qreg.x = half ? qp[2] : qp[0];
    qreg.y = half ? 0.f   : qp[1];     // K=3 pad / K=1

    // ---- pass 1: global row max (per-lane, row = l16) ----
    float mloc = -3.0e38f;
    for (int kt = 0; kt < 64; ++kt) {
      const float2 kk2 = *(const float2*)(kb + (kt * 16 + l16) * 4 + half * 2);
      v2f ka; ka.x = kk2.x; ka.y = kk2.y;   // A = K tile (rows = k-cols)
      v8f c = {};
      c = wmma4(ka, qreg, c);               // c[r] = S[l16][kt*16 + r + half*8]
#pragma unroll
      for (int r = 0; r < 8; ++r) mloc = fmaxf(mloc, c[r]);
    }
    const float m = fmaxf(mloc, __shfl_xor(mloc, 16));

    // ---- pass 2: exp, row sum, O^T accumulation ----
    float lloc = 0.f;
    v8f oc = {};                            // O^T: lane l16 = q-row, c[r] = v-dim r
    for (int kt = 0; kt < 64; ++kt) {
      const float2 kk2 = *(const float2*)(kb + (kt * 16 + l16) * 4 + half * 2);
      v2f ka; ka.x = kk2.x; ka.y = kk2.y;
      v8f c = {};
      c = wmma4(ka, qreg, c);

      float e[8];
#pragma unroll
      for (int r = 0; r < 8; ++r) {
        e[r] = __expf(c[r] - m);            // P[l16][kt*16 + r + half*8]
        lloc += e[r];
      }
      // packed cross-half exchange: each bpermute donates a different value
      // per half (half0 receives cols kt*16+{8,9,12,13}; half1 cols {2,3,6,7})
      const float g0 = __shfl_xor(half ? e[0] : e[2], 16);
      const float g1 = __shfl_xor(half ? e[1] : e[3], 16);
      const float g2 = __shfl_xor(half ? e[4] : e[6], 16);
      const float g3 = __shfl_xor(half ? e[5] : e[7], 16);

      // B = P^T chunk rows (rows kk*4+half*2, kk*4+half*2+1) for each K chunk
      float b2x[4], b2y[4];
      b2x[0] = half ? g0   : e[0];  b2y[0] = half ? g1   : e[1];   // rows 0,1 / 2,3
      b2x[1] = half ? g2   : e[4];  b2y[1] = half ? g3   : e[5];   // rows 4,5 / 6,7
      b2x[2] = half ? e[2] : g0;    b2y[2] = half ? e[3] : g1;     // rows 8,9 / 10,11
      b2x[3] = half ? e[6] : g2;    b2y[3] = half ? e[7] : g3;     // rows 12,13 / 14,15
#pragma unroll
      for (int kk = 0; kk < 4; ++kk) {
        const int k0 = kt * 16 + kk * 4 + half * 2;
        const float2 va = *(const float2*)(vt + (l16 & 3) * VTS + k0);
        v2f a2, b2;                        // A = V^T chunk (rows = v-dims)
        a2.x = (l16 < 4) ? va.x : 0.f;     // row3 of vt is zeros
        a2.y = (l16 < 4) ? va.y : 0.f;
        b2.x = b2x[kk];
        b2.y = b2y[kk];
        oc = wmma4(a2, b2, oc);
      }
    }
    const float lsum = lloc + __shfl_xor(lloc, 16);

    // O^T rows 0..2 live on half0 lanes; lane l16 = q-row
    if (half == 0) {
      const float inv = 1.f / lsum;
      float* op = o + base + (qt * 16 + l16) * EDIM;
      op[0] = oc[0] * inv;
      op[1] = oc[1] * inv;
      op[2] = oc[2] * inv;
    }
  }
}

// ---------------------------------------------------------------------------
// Kernel D: fused MLP head.  h = gelu(sum_p o_p @ FW_p^T + fb); out = tanh(h@w2^T+b2)
// ---------------------------------------------------------------------------
__global__ __launch_bounds__(128) void fuse_kernel(
    const float* __restrict__ o, const float* __restrict__ FW,
    const float* __restrict__ fb, const float* __restrict__ w2,
    const float* __restrict__ b2, float* __restrict__ out) {
  __shared__ float sFW[NP * 48 * EDIM];  // 4608
  __shared__ float sfb[48];
  __shared__ float sw2[EDIM * 48];       // 576
  __shared__ float sb2[EDIM];
  __shared__ float hbuf[128 * 49];       // padded stride 49

  const int tid = threadIdx.x;
  for (int i = tid; i < NP * 48 * EDIM; i += 128) sFW[i] = FW[i];
  for (int i = tid; i < 48; i += 128) sfb[i] = fb[i];
  for (int i = tid; i < EDIM * 48; i += 128) sw2[i] = w2[i];
  for (int i = tid; i < EDIM; i += 128) sb2[i] = b2[i];
  __syncthreads();

  const int t = blockIdx.x * 128 + tid;  // token 0..4095
  float ovr[NP * EDIM];
#pragma unroll
  for (int p = 0; p < NP; ++p)
#pragma unroll
    for (int e = 0; e < EDIM; ++e)
      ovr[p * EDIM + e] = o[p * (NTOK * EDIM) + t * EDIM + e];

  for (int f = 0; f < 48; ++f) {
    float acc = sfb[f];
#pragma unroll
    for (int p = 0; p < NP; ++p)
#pragma unroll
      for (int e = 0; e < EDIM; ++e)
        acc += ovr[p * EDIM + e] * sFW[p * (48 * EDIM) + f * EDIM + e];
    // exact gelu: x * 0.5 * (1 + erf(x/sqrt(2)))
    hbuf[tid * 49 + f] = 0.5f * acc * (1.f + erff(acc * 0.70710678118654752f));
  }

  for (int f2 = 0; f2 < EDIM; ++f2) {
    float acc = sb2[f2];
    for (int g = 0; g < 48; ++g)
      acc += hbuf[tid * 49 + g] * sw2[f2 * 48 + g];
    out[t * EDIM + f2] = tanhf(acc);
  }
}

// ---------------------------------------------------------------------------
extern "C" void kernel_launch(void* const* d_in, const int* in_sizes, int n_in,
                              void* d_out, int out_size, void* d_ws, size_t ws_size,
                              hipStream_t stream) {
  (void)in_sizes; (void)n_in; (void)out_size; (void)ws_size;
  const float* x      = (const float*)d_in[0];
  const float* proj_w = (const float*)d_in[1];
  const float* proj_b = (const float*)d_in[2];
  const float* in_w   = (const float*)d_in[3];
  const float* in_b   = (const float*)d_in[4];
  const float* out_w  = (const float*)d_in[5];
  const float* out_b  = (const float*)d_in[6];
  const float* fus_w1 = (const float*)d_in[7];
  const float* fus_b1 = (const float*)d_in[8];
  const float* fus_w2 = (const float*)d_in[9];
  const float* fus_b2 = (const float*)d_in[10];
  float* out = (float*)d_out;

  float* ws   = (float*)d_ws;
  float* q    = ws;
  float* k    = q + QKVSZ;
  float* v    = k + QKVSZ;
  float* o    = v + QKVSZ;
  float* Wqkv = o + QKVSZ;      // 3456
  float* bqkv = Wqkv + 3456;    // 288
  float* FW   = bqkv + 288;     // 4608
  float* fb   = FW + 4608;      // 48

  fold_kernel<<<1, 256, 0, stream>>>(proj_w, proj_b, in_w, in_b, out_w, out_b,
                                     fus_w1, fus_b1, Wqkv, bqkv, FW, fb);
  qkv_kernel<<<dim3(NTOK / 256, NP), 256, 0, stream>>>(x, Wqkv, bqkv, q, k, v);
  attn_kernel<<<NP * NB * NH, 256, 0, stream>>>(q, k, v, o);
  fuse_kernel<<<NTOK / 128, 128, 0, stream>>>(o, FW, fb, fus_w2, fus_b2, out);
}